// GAT_layer_21534966022608
// MI455X (gfx1250) — compile-verified
//
#include <hip/hip_runtime.h>
#include <math.h>

// GAT layer, CDNA5 (gfx1250, wave32).
// Math: reference segment-sums exp_e AND contrib by `src`, so
//   h_new[n,h,:] = h_proj[n,h,:] * denom[n,h] / (denom[n,h] + 1e-16)
// which removes the 51M-atomic contrib scatter entirely.

typedef float v2f __attribute__((ext_vector_type(2)));
typedef float v8f __attribute__((ext_vector_type(8)));

#define F_IN   128
#define HF     32   // H * F_OUT
#define NHEAD  4
#define FOUT   8

// ---------------------------------------------------------------------------
// Kernel 1: h_proj = h_in @ W^T + b  via V_WMMA_F32_16X16X4_F32.
// One wave computes a 16-row x 32-col tile with two accumulators.
// A 16x4 f32 tile: lane l -> row (l&15), K = 2*(l>>4) + {0,1}  (float2 load)
// B 4x16 f32 tile: lane l -> col (l&15), K = 2*(l>>4) + {0,1}  (float2 load)
// D 16x16 f32:     VGPR r -> row r + 8*(l>>4), col (l&15)
// ---------------------------------------------------------------------------
__global__ __launch_bounds__(256) void gat_proj_wmma(
    const float* __restrict__ h_in, const float* __restrict__ W,
    const float* __restrict__ bias, float* __restrict__ h_proj, int N)
{
    const int gwave = (blockIdx.x * blockDim.x + threadIdx.x) >> 5;
    const int lane  = threadIdx.x & 31;
    const int row0  = gwave * 16;
    if (row0 >= N) return;                 // uniform per wave: EXEC stays full

    const int rA = lane & 15;              // A row within tile
    const int kk = (lane >> 4) * 2;        // K sub-offset for this lane half
    int rowg = row0 + rA;
    if (rowg >= N) rowg = N - 1;           // clamp loads; stores are guarded

    const float* __restrict__ Ap = h_in + (size_t)rowg * F_IN + kk;
    const int nB = lane & 15;              // B column within tile
    const float* __restrict__ B0 = W + (size_t)nB        * F_IN + kk; // cols 0..15
    const float* __restrict__ B1 = W + (size_t)(nB + 16) * F_IN + kk; // cols 16..31

    v8f c0 = {};
    v8f c1 = {};
    #pragma unroll
    for (int k0 = 0; k0 < F_IN; k0 += 4) {
        v2f a  = *(const v2f*)(Ap + k0);
        v2f b0 = *(const v2f*)(B0 + k0);
        v2f b1 = *(const v2f*)(B1 + k0);
        c0 = __builtin_amdgcn_wmma_f32_16x16x4_f32(
                 false, a, false, b0, (short)0, c0, false, false);
        c1 = __builtin_amdgcn_wmma_f32_16x16x4_f32(
                 false, a, false, b1, (short)0, c1, false, false);
    }

    const int col   = lane & 15;
    const int rbase = (lane >> 4) * 8;
    const float bia0 = bias[col];
    const float bia1 = bias[col + 16];
    #pragma unroll
    for (int r = 0; r < 8; ++r) {
        const int m = row0 + rbase + r;
        if (m < N) {
            h_proj[(size_t)m * HF + col]      = c0[r] + bia0;
            h_proj[(size_t)m * HF + col + 16] = c1[r] + bia1;
        }
    }
}

// ---------------------------------------------------------------------------
// Kernel 2: per-node attention scores + zero denom (fresh every call).
// ---------------------------------------------------------------------------
__global__ __launch_bounds__(256) void gat_scores(
    const float* __restrict__ h_proj,
    const float* __restrict__ a_src, const float* __restrict__ a_tgt,
    float* __restrict__ s_src, float* __restrict__ s_tgt,
    float* __restrict__ denom, int N)
{
    const int n = blockIdx.x * blockDim.x + threadIdx.x;
    if (n >= N) return;

    const float4* hp = (const float4*)(h_proj + (size_t)n * HF);
    float4 ss, st;
    float* pss = (float*)&ss;
    float* pst = (float*)&st;
    #pragma unroll
    for (int h = 0; h < NHEAD; ++h) {
        const float4 x0 = hp[h * 2 + 0];
        const float4 x1 = hp[h * 2 + 1];
        const float* as = a_src + h * FOUT;
        const float* at = a_tgt + h * FOUT;
        pss[h] = x0.x * as[0] + x0.y * as[1] + x0.z * as[2] + x0.w * as[3]
               + x1.x * as[4] + x1.y * as[5] + x1.z * as[6] + x1.w * as[7];
        pst[h] = x0.x * at[0] + x0.y * at[1] + x0.z * at[2] + x0.w * at[3]
               + x1.x * at[4] + x1.y * at[5] + x1.z * at[6] + x1.w * at[7];
    }
    ((float4*)s_src)[n] = ss;
    ((float4*)s_tgt)[n] = st;
    float4 z; z.x = 0.f; z.y = 0.f; z.z = 0.f; z.w = 0.f;
    ((float4*)denom)[n] = z;
}

// ---------------------------------------------------------------------------
// Kernel 3: per-edge softmax-denominator accumulation (L2-resident atomics).
// ---------------------------------------------------------------------------
__global__ __launch_bounds__(256) void gat_edges(
    const int* __restrict__ edge_index,
    const float* __restrict__ s_src, const float* __restrict__ s_tgt,
    float* __restrict__ denom, int E)
{
    const int e = blockIdx.x * blockDim.x + threadIdx.x;
    if (e >= E) return;
    const int src = edge_index[e];
    const int tgt = edge_index[(size_t)E + e];

    float4 ss = ((const float4*)s_src)[src];
    float4 st = ((const float4*)s_tgt)[tgt];
    const float* p0 = (const float*)&ss;
    const float* p1 = (const float*)&st;
    #pragma unroll
    for (int h = 0; h < NHEAD; ++h) {
        float v = p0[h] + p1[h];
        v = (v > 0.0f) ? v : 0.2f * v;      // LeakyReLU(0.2)
        atomicAdd(&denom[(size_t)src * NHEAD + h], __expf(v));
    }
}

// ---------------------------------------------------------------------------
// Kernel 4: h_new = h_proj * denom/(denom + 1e-16)
// ---------------------------------------------------------------------------
__global__ __launch_bounds__(256) void gat_finalize(
    const float* __restrict__ h_proj, const float* __restrict__ denom,
    float* __restrict__ out, int N)
{
    const int n = blockIdx.x * blockDim.x + threadIdx.x;
    if (n >= N) return;

    const float4 d = ((const float4*)denom)[n];
    float sc[NHEAD];
    sc[0] = d.x / (d.x + 1e-16f);
    sc[1] = d.y / (d.y + 1e-16f);
    sc[2] = d.z / (d.z + 1e-16f);
    sc[3] = d.w / (d.w + 1e-16f);

    const float4* hp = (const float4*)(h_proj + (size_t)n * HF);
    float4*       op = (float4*)(out + (size_t)n * HF);
    #pragma unroll
    for (int i = 0; i < 8; ++i) {
        const float s = sc[i >> 1];
        float4 x = hp[i];
        x.x *= s; x.y *= s; x.z *= s; x.w *= s;
        op[i] = x;
    }
}

// ---------------------------------------------------------------------------
extern "C" void kernel_launch(void* const* d_in, const int* in_sizes, int n_in,
                              void* d_out, int out_size, void* d_ws, size_t ws_size,
                              hipStream_t stream)
{
    const float* h_in  = (const float*)d_in[0];
    const float* W     = (const float*)d_in[1];
    const float* bias  = (const float*)d_in[2];
    const float* a_src = (const float*)d_in[3];
    const float* a_tgt = (const float*)d_in[4];
    const int*   eidx  = (const int*)d_in[5];

    const int N = in_sizes[0] / F_IN;
    const int E = in_sizes[5] / 2;

    float* out = (float*)d_out;

    // Workspace layout (floats): h_proj | s_src | s_tgt | denom
    float* h_proj = (float*)d_ws;
    float* s_src  = h_proj + (size_t)N * HF;
    float* s_tgt  = s_src  + (size_t)N * NHEAD;
    float* denom  = s_tgt  + (size_t)N * NHEAD;

    // 1) WMMA projection: 8 waves/block, 16 rows/wave -> 128 rows/block
    const int rowsPerBlock = 128;
    const int gridProj = (N + rowsPerBlock - 1) / rowsPerBlock;
    gat_proj_wmma<<<gridProj, 256, 0, stream>>>(h_in, W, bias, h_proj, N);

    // 2) Scores + denom zeroing
    const int gridN = (N + 255) / 256;
    gat_scores<<<gridN, 256, 0, stream>>>(h_proj, a_src, a_tgt,
                                          s_src, s_tgt, denom, N);

    // 3) Edge accumulation
    const int gridE = (E + 255) / 256;
    gat_edges<<<gridE, 256, 0, stream>>>(eidx, s_src, s_tgt, denom, E);

    // 4) Finalize
    gat_finalize<<<gridN, 256, 0, stream>>>(h_proj, denom, out, N);
}